// JointEntropy_13434657702285
// MI455X (gfx1250) — compile-verified
//
#include <hip/hip_runtime.h>

typedef float v2f __attribute__((ext_vector_type(2)));
typedef float v8f __attribute__((ext_vector_type(8)));

#define Hh     192
#define Ww     192
#define HO     190
#define WO     190
#define HWsz   (Hh * Ww)
#define FRAME  (3 * HWsz)
#define PITCH  12            // floats per Gram row (48B -> 16B aligned)

// H = ln(M) + 0.5*C*ln(2*pi*h^2) - mean_i ln s_i ; h=0.1, M=9
#define CONST_MARG  (-1.9537151021f)     // ln9 + 1.5*ln(2*pi*0.01)
#define CONST_JOINT (-6.1046547815f)     // ln9 + 3.0*ln(2*pi*0.01)
#define NEG50_LOG2E (-72.13475204444817f)  // -50 * log2(e)
#define POS100_LOG2E (144.26950408889634f) // +100 * log2(e)
#define LN2_OVER_9  (0.07701635339554946f) // ln2 / 9

__device__ __forceinline__ float rlane(float v, int l) {
    return __int_as_float(__builtin_amdgcn_readlane(__float_as_int(v), l));
}

__global__ __launch_bounds__(256) void je_zero(float* __restrict__ out, int n) {
    int i = blockIdx.x * 256 + threadIdx.x;
    if (i < n) out[i] = 0.0f;
}

__global__ __launch_bounds__(256) void je_kernel(const float* __restrict__ in,
                                                 float* __restrict__ out) {
    // per-wave scratch: two 9x9 Grams, pitch 12, 112 floats each (16B aligned)
    __shared__ float4 ldsv[8][2][28];

    const int tid  = threadIdx.x;
    const int wave = tid >> 5;
    const int lane = tid & 31;
    const int y    = blockIdx.x;      // patch row 0..189
    const int nsf  = blockIdx.y;      // flattened (n, sf) 0..15
    const int sf   = nsf & 3;
    const int sfn  = (sf == 3) ? sf : sf + 1;
    const int curBase = nsf * FRAME;
    const int nxtBase = ((nsf & ~3) + sfn) * FRAME;

    // A-matrix (f32 16x4) lane layout: lanes 0-15 hold M=lane, K=0..1;
    // lanes 16-31 hold M=lane-16, K=2..3.
    const int  m  = lane & 15;            // patch-point row (valid: 0..8)
    const int  kb = (lane >> 4) * 2;      // channel base: 0 or 2
    const bool vm = (m < 9);
    const bool vy = vm && (lane < 16);    // channel kb+1 in range (1; 3 -> pad)
    const int  dy = m / 3;
    const int  dx = m - dy * 3;
    const int  pixBase = (y + dy) * Ww + dx;

    // flat element-index bases; x added per iteration, clamped for pad lanes
    const int ixc = curBase + kb * HWsz + pixBase;
    const int iyc = curBase + (kb + 1) * HWsz + pixBase;
    const int ixn = nxtBase + kb * HWsz + pixBase;
    const int iyn = nxtBase + (kb + 1) * HWsz + pixBase;

    float* ldsC = (float*)&ldsv[wave][0][0];
    float* ldsN = (float*)&ldsv[wave][1][0];

    const int   i9 = vm ? m : 8;                 // clamped row id (uniform exec)
    const float wj = (lane < 16) ? 0.0f : 1.0f;  // joint-term weight

    // hoisted LDS pointers
    float4* stC = (float4*)(ldsC + m * PITCH);     // store: column m == row m (symmetry)
    float4* stN = (float4*)(ldsN + m * PITCH);
    const float4* rdC = (const float4*)(ldsC + i9 * PITCH);
    const float4* rdN = (const float4*)(ldsN + i9 * PITCH);

    for (int x = wave; x < WO; x += 8) {          // wave-uniform loop
        // ---- gather A operands (branch-free, clamped addresses + value masks)
        float t0 = in[vm ? (ixc + x) : 0];
        float t1 = in[vy ? (iyc + x) : 0];
        float t2 = in[vm ? (ixn + x) : 0];
        float t3 = in[vy ? (iyn + x) : 0];
        v2f ac, an;
        ac.x = vm ? t0 : 0.0f;
        ac.y = vy ? t1 : 0.0f;
        an.x = vm ? t2 : 0.0f;
        an.y = vy ? t3 : 0.0f;

        // ---- Gram = X * X^T (A/B register images coincide for symmetric product)
        v8f z = {0.f, 0.f, 0.f, 0.f, 0.f, 0.f, 0.f, 0.f};
        v8f gc = __builtin_amdgcn_wmma_f32_16x16x4_f32(false, ac, false, ac,
                                                       (short)0, z, false, false);
        v8f gn = __builtin_amdgcn_wmma_f32_16x16x4_f32(false, an, false, an,
                                                       (short)0, z, false, false);

        // ---- scatter Gram to LDS (C-layout: lane<16 c[v]=G[v][lane]; lane>=16 c[v]=G[v+8][lane-16])
        if (vm && lane < 16) {
            stC[0] = make_float4(gc[0], gc[1], gc[2], gc[3]);
            stC[1] = make_float4(gc[4], gc[5], gc[6], gc[7]);
            stN[0] = make_float4(gn[0], gn[1], gn[2], gn[3]);
            stN[1] = make_float4(gn[4], gn[5], gn[6], gn[7]);
        }
        if (vm && lane >= 16) {
            ldsC[m * PITCH + 8] = gc[0];   // row-8 entries (incl. G[8][8] via m=8)
            ldsN[m * PITCH + 8] = gn[0];
        }
        __builtin_amdgcn_wave_barrier();
        asm volatile("s_wait_dscnt 0" ::: "memory");
        __builtin_amdgcn_wave_barrier();

        // ---- branch-free entropy rows: lanes 0-8 marginal (wj=0), 16-24 joint (wj=1)
        float4 rc0 = rdC[0];
        float4 rc1 = rdC[1];
        float  rc8 = ldsC[i9 * PITCH + 8];
        float4 rn0 = rdN[0];
        float4 rn1 = rdN[1];
        float  rn8 = ldsN[i9 * PITCH + 8];
        // own diagonals, pre-scaled by -50*log2(e) so the j-loop feeds exp2 directly
        float  sqc = ldsC[i9 * 13] * NEG50_LOG2E;   // PITCH*i + i
        float  sqn = ldsN[i9 * 13] * NEG50_LOG2E;

        const float rc[9] = {rc0.x, rc0.y, rc0.z, rc0.w, rc1.x, rc1.y, rc1.z, rc1.w, rc8};
        const float rn[9] = {rn0.x, rn0.y, rn0.z, rn0.w, rn1.x, rn1.y, rn1.z, rn1.w, rn8};

        float s = 0.0f;
#pragma unroll
        for (int j = 0; j < 9; ++j) {
            // uniform broadcasts: v_readlane -> SGPR (lanes 0-8 carry sqc_j, 16-24 sqn_j)
            float sqcj = rlane(sqc, j);
            float sqnj = rlane(sqn, 16 + j);
            // t~ = -50*log2e * (sq_i + sq_j - 2*G_ij)   (<= 0 after clamp)
            float tc = fmaf(POS100_LOG2E, rc[j], sqc + sqcj);
            float tn = fmaf(POS100_LOG2E, rn[j], sqn + sqnj);
            float u  = fminf(fmaf(wj, tn, tc), 0.0f);
            s += __builtin_amdgcn_exp2f(u);           // v_exp_f32
        }
        // raw v_log_f32 (base-2); ln2 folded into the final scale
        float acc = vm ? __builtin_amdgcn_logf(s) : 0.0f;
        __builtin_amdgcn_wave_barrier();

        // half-wave tree reduce: lane0 = sum(lanes 0..15), lane16 = sum(16..31)
#pragma unroll
        for (int off = 8; off > 0; off >>= 1) acc += __shfl_down(acc, off);

        if (lane == 0)
            out[(nsf * 2 + 0) * HWsz + (y + 1) * Ww + (x + 1)] =
                CONST_MARG - acc * LN2_OVER_9;
        if (lane == 16)
            out[(nsf * 2 + 1) * HWsz + (y + 1) * Ww + (x + 1)] =
                CONST_JOINT - acc * LN2_OVER_9;
    }
}

extern "C" void kernel_launch(void* const* d_in, const int* in_sizes, int n_in,
                              void* d_out, int out_size, void* d_ws, size_t ws_size,
                              hipStream_t stream) {
    const float* in  = (const float*)d_in[0];
    float*       out = (float*)d_out;

    // zero entire output (border stays zero; interior overwritten below)
    je_zero<<<(out_size + 255) / 256, 256, 0, stream>>>(out, out_size);

    dim3 grid(HO, 16);   // 190 patch rows x (N*SF = 16)
    je_kernel<<<grid, 256, 0, stream>>>(in, out);
}